// MultiHeadAttentionQuantum_65481071397430
// MI455X (gfx1250) — compile-verified
//
#include <hip/hip_runtime.h>
#include <stdint.h>

// Analytic reduction of the 8-qubit RX + CNOT-ring circuit:
//   out[g*8+0] = prod_{k=1..7} cos(x[g*8+k] + theta[k])
//   out[g*8+j] = prod_{k=0..j} cos(x[g*8+k] + theta[k]),  j = 1..7
// (exact: CNOTs permute basis states, so probabilities stay a product
//  distribution over independent bits; E[(-1)^XOR] factorizes into
//  cos^2(phi/2) - sin^2(phi/2) = cos(phi) per qubit.)
//
// Memory-bound streaming kernel: 2MB in + 2MB out.
// Input is staged via the CDNA5 async data mover (global->LDS, ASYNCcnt),
// the canonical gfx1250 bulk-input path (no load-return VGPR pressure).

#define TPB 128   // 4 wave32 per block; 1 group of 8 floats per thread

__global__ __launch_bounds__(TPB) void quantum_rx_ring_kernel(
    const float* __restrict__ x,
    const float* __restrict__ theta,
    float* __restrict__ out,
    int ngroups)
{
    __shared__ __align__(16) float tile[TPB * 8];   // 4 KB staging buffer

    const int tid = (int)threadIdx.x;
    const int g   = (int)blockIdx.x * TPB + tid;

    // theta is uniform with constant indices -> scalar (s_load) path.
    const float t0 = theta[0], t1 = theta[1], t2 = theta[2], t3 = theta[3];
    const float t4 = theta[4], t5 = theta[5], t6 = theta[6], t7 = theta[7];

    // Raw LDS byte offset: flat-address low 32 bits == wave-relative LDS
    // address (LDS aperture lives entirely in ADDR[63:32], ISA sec 10.2).
    const unsigned lds_off =
        (unsigned)(uintptr_t)(const void*)&tile[tid * 8];
    const unsigned long long gaddr =
        (unsigned long long)(uintptr_t)(x + (size_t)g * 8);

    if (g < ngroups) {
        // 32 bytes per lane, two B128 async transfers.
        // IOFFSET applies to BOTH the LDS and global addresses
        asm volatile("global_load_async_to_lds_b128 %0, %1, off"
                     :: "v"(lds_off), "v"(gaddr) : "memory");
        asm volatile("global_load_async_to_lds_b128 %0, %1, off offset:16"
                     :: "v"(lds_off), "v"(gaddr) : "memory");
    }

    // Each lane reads back only the region IT loaded -> waiting on this
    // wave's ASYNCcnt is sufficient; no cross-wave barrier needed.
    asm volatile("s_wait_asynccnt 0x0" ::: "memory");

    if (g >= ngroups) return;

    const float4* f4 = reinterpret_cast<const float4*>(&tile[tid * 8]);
    const float4 a = f4[0];
    const float4 b = f4[1];

    const float c0 = cosf(a.x + t0);
    const float c1 = cosf(a.y + t1);
    const float c2 = cosf(a.z + t2);
    const float c3 = cosf(a.w + t3);
    const float c4 = cosf(b.x + t4);
    const float c5 = cosf(b.y + t5);
    const float c6 = cosf(b.z + t6);
    const float c7 = cosf(b.w + t7);

    // Prefix products p_j = c0*...*cj  (outputs 1..7)
    const float p1 = c0 * c1;
    const float p2 = p1 * c2;
    const float p3 = p2 * c3;
    const float p4 = p3 * c4;
    const float p5 = p4 * c5;
    const float p6 = p5 * c6;
    const float p7 = p6 * c7;
    // Suffix product c1*...*c7 (output 0) -- no division by c0 (may be ~0).
    const float s  = ((c1 * c2) * (c3 * c4)) * ((c5 * c6) * c7);

    float4* op = reinterpret_cast<float4*>(out + (size_t)g * 8);
    op[0] = make_float4(s,  p1, p2, p3);
    op[1] = make_float4(p4, p5, p6, p7);
}

extern "C" void kernel_launch(void* const* d_in, const int* in_sizes, int n_in,
                              void* d_out, int out_size, void* d_ws, size_t ws_size,
                              hipStream_t stream) {
    const float* x     = (const float*)d_in[0];   // (2, 512, 512) f32
    const float* theta = (const float*)d_in[1];   // (8,) f32
    float* out         = (float*)d_out;           // (2, 512, 512) f32

    const int ngroups = in_sizes[0] / 8;          // 65536 head-groups
    const int blocks  = (ngroups + TPB - 1) / TPB;

    quantum_rx_ring_kernel<<<blocks, TPB, 0, stream>>>(x, theta, out, ngroups);
}